// GAT_44203803410475
// MI455X (gfx1250) — compile-verified
//
#include <hip/hip_runtime.h>

// GATv2 x2 + BN for MI455X (gfx1250), wave32.
// GEMMs via V_WMMA_F32_16X16X4_F32 (exact f32 numerics, matrix pipe).
// Edge softmax/aggregation via L2-resident gathers + atomics.

typedef float v2f __attribute__((ext_vector_type(2)));
typedef float v8f __attribute__((ext_vector_type(8)));

#define SLOPE_ATT 0.2f
#define SLOPE_OUT 0.01f
#define BN_EPS 1e-5f

static inline int cdiv_i(long a, long b) { return (int)((a + b - 1) / b); }

// ---- float <-> order-preserving unsigned (for atomicMax on float) ----
__device__ __forceinline__ unsigned enc_f32(float f) {
  unsigned b = __float_as_uint(f);
  return (b & 0x80000000u) ? ~b : (b | 0x80000000u);
}
__device__ __forceinline__ float dec_f32(unsigned u) {
  return (u & 0x80000000u) ? __uint_as_float(u & 0x7FFFFFFFu) : __uint_as_float(~u);
}

// ---------------- fill helpers ----------------
__global__ void fill_f32_k(float* p, float v, long n) {
  long i = (long)blockIdx.x * blockDim.x + threadIdx.x;
  long s = (long)gridDim.x * blockDim.x;
  for (; i < n; i += s) p[i] = v;
}
__global__ void fill_u32_k(unsigned* p, unsigned v, long n) {
  long i = (long)blockIdx.x * blockDim.x + threadIdx.x;
  long s = (long)gridDim.x * blockDim.x;
  for (; i < n; i += s) p[i] = v;
}

// ---------------- dual GEMM: OL = X@WL + bL, OR = X@WR + bR ----------------
// One wave per 16x16 output tile. K stepped by 4 using V_WMMA_F32_16X16X4_F32.
// A (16x4, MxK): lanes 0-15 hold M rows; VGPR0 = K{0 | 2}, VGPR1 = K{1 | 3} by lane half.
// B (4x16, KxN): lane half selects K pair {0,1} vs {2,3}; VGPR selects K in pair; lane&15 = N.
// C/D (16x16): VGPR r, lane half 0 -> M=r, half 1 -> M=r+8; N = lane&15.
__global__ void gemm_dual_wmma_k(const float* __restrict__ X,
                                 const float* __restrict__ WL, const float* __restrict__ bL,
                                 const float* __restrict__ WR, const float* __restrict__ bR,
                                 float* __restrict__ OL, float* __restrict__ OR_,
                                 int M, int K, int Nout) {
  int wave = threadIdx.x >> 5;
  int lane = threadIdx.x & 31;
  int ntiles = Nout >> 4;
  int mtiles = (M + 15) >> 4;
  int tile = blockIdx.x * (blockDim.x >> 5) + wave;
  if (tile >= mtiles * ntiles) return;
  int mt = tile / ntiles, nt = tile - mt * ntiles;
  int half = lane >> 4, lr = lane & 15;

  int row = (mt << 4) + lr;
  int rowc = row < M ? row : (M - 1);  // clamp A loads (stores are guarded)
  int col = (nt << 4) + lr;
  const float* __restrict__ xrow = X + (size_t)rowc * K;

  v8f accL = {};
  v8f accR = {};
  for (int k0 = 0; k0 < K; k0 += 4) {
    int ka = k0 + (half << 1);
    v2f a;
    a.x = xrow[ka];
    a.y = xrow[ka + 1];
    size_t o0 = (size_t)ka * Nout + col;
    size_t o1 = o0 + Nout;
    v2f bl_;
    bl_.x = WL[o0];
    bl_.y = WL[o1];
    v2f br_;
    br_.x = WR[o0];
    br_.y = WR[o1];
    accL = __builtin_amdgcn_wmma_f32_16x16x4_f32(false, a, false, bl_, (short)0, accL, false, false);
    accR = __builtin_amdgcn_wmma_f32_16x16x4_f32(false, a, false, br_, (short)0, accR, false, false);
  }
  float addL = bL[col], addR = bR[col];
#pragma unroll
  for (int r = 0; r < 8; r++) {
    int orow = (mt << 4) + r + (half << 3);
    if (orow < M) {
      size_t o = (size_t)orow * Nout + col;
      OL[o] = accL[r] + addL;
      OR_[o] = accR[r] + addR;
    }
  }
}

// ---------------- edge attention logits (wave per edge) ----------------
// logit[e][h] = sum_c att[h*64+c'] * leaky(xl[src][h*64+c'] + xr[dst][h*64+c'], 0.2)
__global__ void edge_logits_k(const int* __restrict__ ei, int E, long Etot,
                              const float* __restrict__ XL, const float* __restrict__ XR,
                              const float* __restrict__ ATT, float* __restrict__ LG) {
  int wave = threadIdx.x >> 5;
  int lane = threadIdx.x & 31;
  long e = (long)blockIdx.x * (blockDim.x >> 5) + wave;
  if (e >= Etot) return;
  int src = (e < E) ? ei[e] : (int)(e - E);
  int dst = (e < E) ? ei[E + e] : (int)(e - E);
  const float* __restrict__ xl = XL + (size_t)src * 256;
  const float* __restrict__ xr = XR + (size_t)dst * 256;
#pragma unroll
  for (int h = 0; h < 4; h++) {
    float s = 0.f;
#pragma unroll
    for (int i = 0; i < 2; i++) {
      int c = (h << 6) + lane + (i << 5);
      float v = xl[c] + xr[c];
      v = v > 0.f ? v : SLOPE_ATT * v;
      s += v * ATT[c];  // att flat [4][64] == channel index
    }
#pragma unroll
    for (int off = 16; off; off >>= 1) s += __shfl_xor(s, off, 32);
    if (lane == 0) LG[e * 4 + h] = s;
  }
}

// ---------------- segment max over dst (thread per edge-head) ----------------
__global__ void seg_max_k(const int* __restrict__ ei, int E, long Etot,
                          const float* __restrict__ LG, unsigned* __restrict__ MX) {
  long idx = (long)blockIdx.x * blockDim.x + threadIdx.x;
  if (idx >= Etot * 4) return;
  long e = idx >> 2;
  int h = (int)(idx & 3);
  int dst = (e < E) ? ei[E + e] : (int)(e - E);
  atomicMax(&MX[(size_t)dst * 4 + h], enc_f32(LG[idx]));
}

// ---------------- exp(logit - max) and segment denom ----------------
__global__ void seg_exp_k(const int* __restrict__ ei, int E, long Etot,
                          float* __restrict__ LG, const unsigned* __restrict__ MX,
                          float* __restrict__ DEN) {
  long idx = (long)blockIdx.x * blockDim.x + threadIdx.x;
  if (idx >= Etot * 4) return;
  long e = idx >> 2;
  int h = (int)(idx & 3);
  int dst = (e < E) ? ei[E + e] : (int)(e - E);
  float p = expf(LG[idx] - dec_f32(MX[(size_t)dst * 4 + h]));
  LG[idx] = p;
  atomicAdd(&DEN[(size_t)dst * 4 + h], p);
}

// ---------------- weighted aggregation (wave per edge) ----------------
__global__ void aggregate_k(const int* __restrict__ ei, int E, long Etot,
                            const float* __restrict__ XL, const float* __restrict__ LG,
                            const float* __restrict__ DEN, float* __restrict__ OUT) {
  int wave = threadIdx.x >> 5;
  int lane = threadIdx.x & 31;
  long e = (long)blockIdx.x * (blockDim.x >> 5) + wave;
  if (e >= Etot) return;
  int src = (e < E) ? ei[e] : (int)(e - E);
  int dst = (e < E) ? ei[E + e] : (int)(e - E);
  float alpha_l = 0.f;
  if (lane < 4) alpha_l = LG[e * 4 + lane] / (DEN[(size_t)dst * 4 + lane] + 1e-16f);
  const float* __restrict__ xl = XL + (size_t)src * 256;
  float* __restrict__ o = OUT + (size_t)dst * 256;
#pragma unroll
  for (int i = 0; i < 8; i++) {
    int c = lane + (i << 5);
    float a = __shfl(alpha_l, c >> 6, 32);
    atomicAdd(&o[c], xl[c] * a);
  }
}

// ---- bias + leaky(0.01) in place, accumulate per-column sum / sumsq ----
// blockDim.x == C (number of columns); each block handles a row chunk, coalesced.
__global__ void bias_leaky_stats_k(float* __restrict__ H, const float* __restrict__ BIAS,
                                   float* __restrict__ SUM, float* __restrict__ SQ,
                                   int Nrows, int C, int rowsPerBlock) {
  int col = threadIdx.x;
  int r0 = blockIdx.x * rowsPerBlock;
  int r1 = r0 + rowsPerBlock;
  if (r1 > Nrows) r1 = Nrows;
  float b = BIAS[col];
  float s = 0.f, q = 0.f;
  for (int r = r0; r < r1; r++) {
    size_t o = (size_t)r * C + col;
    float v = H[o] + b;
    v = v > 0.f ? v : SLOPE_OUT * v;
    H[o] = v;
    s += v;
    q += v * v;
  }
  atomicAdd(&SUM[col], s);
  atomicAdd(&SQ[col], q);
}

// ---- head mean (4 heads x 64) + bias + leaky + stats, for layer-2 output ----
__global__ void head_mean_stats_k(const float* __restrict__ AGG, const float* __restrict__ BIAS,
                                  float* __restrict__ H2, float* __restrict__ SUM,
                                  float* __restrict__ SQ, int Nrows, int rowsPerBlock) {
  int col = threadIdx.x;  // blockDim.x == 64
  int r0 = blockIdx.x * rowsPerBlock;
  int r1 = r0 + rowsPerBlock;
  if (r1 > Nrows) r1 = Nrows;
  float b = BIAS[col];
  float s = 0.f, q = 0.f;
  for (int r = r0; r < r1; r++) {
    const float* a = AGG + (size_t)r * 256;
    float v = 0.25f * (a[col] + a[64 + col] + a[128 + col] + a[192 + col]) + b;
    v = v > 0.f ? v : SLOPE_OUT * v;
    H2[(size_t)r * 64 + col] = v;
    s += v;
    q += v * v;
  }
  atomicAdd(&SUM[col], s);
  atomicAdd(&SQ[col], q);
}

// ---- BN normalize: out = g*(h-mu)*rsqrt(var+eps)+beta (biased var) ----
__global__ void bn_norm_k(const float* __restrict__ H, float* __restrict__ OUT,
                          const float* __restrict__ SUM, const float* __restrict__ SQ,
                          const float* __restrict__ G, const float* __restrict__ BETA,
                          int Nrows, int C) {
  long n = (long)Nrows * C;
  long i = (long)blockIdx.x * blockDim.x + threadIdx.x;
  long stride = (long)gridDim.x * blockDim.x;
  float invN = 1.0f / (float)Nrows;
  for (; i < n; i += stride) {
    int col = (int)(i % C);
    float mu = SUM[col] * invN;
    float var = SQ[col] * invN - mu * mu;
    OUT[i] = G[col] * (H[i] - mu) * rsqrtf(var + BN_EPS) + BETA[col];
  }
}

extern "C" void kernel_launch(void* const* d_in, const int* in_sizes, int n_in,
                              void* d_out, int out_size, void* d_ws, size_t ws_size,
                              hipStream_t stream) {
  const float* x = (const float*)d_in[0];
  const int* ei = (const int*)d_in[1];
  const float* Wl1 = (const float*)d_in[2];
  const float* bl1 = (const float*)d_in[3];
  const float* Wr1 = (const float*)d_in[4];
  const float* br1 = (const float*)d_in[5];
  const float* att1 = (const float*)d_in[6];
  const float* bias1 = (const float*)d_in[7];
  const float* Wl2 = (const float*)d_in[8];
  const float* bl2 = (const float*)d_in[9];
  const float* Wr2 = (const float*)d_in[10];
  const float* br2 = (const float*)d_in[11];
  const float* att2 = (const float*)d_in[12];
  const float* bias2 = (const float*)d_in[13];
  const float* g1 = (const float*)d_in[14];
  const float* beta1 = (const float*)d_in[15];
  const float* g2 = (const float*)d_in[16];
  const float* beta2 = (const float*)d_in[17];
  float* out = (float*)d_out;

  const int N = in_sizes[0] / 128;
  const int E = in_sizes[1] / 2;
  const long Etot = (long)E + N;

  // ---- workspace carve ----
  float* A = (float*)d_ws;               // N x 256  (xl)
  float* B = A + (size_t)N * 256;        // N x 256  (xr; later reused as N x 64 pre-BN)
  float* C = B + (size_t)N * 256;        // N x 256  (layer output / aggregation target)
  float* LG = C + (size_t)N * 256;       // Etot x 4 (logits -> exp p)
  unsigned* MX = (unsigned*)(LG + Etot * 4);  // N x 4 (encoded max)
  float* DEN = (float*)(MX + (size_t)N * 4);  // N x 4
  float* SUM = DEN + (size_t)N * 4;      // 256
  float* SQ = SUM + 256;                 // 256
  float* H2 = B;                         // N x 64 (layer-2 pre-BN), reuse of xr space

  const unsigned ENC_NEG_INF = 0x007FFFFFu;  // enc_f32(-inf) = ~0xFF800000
  const int mtiles = (N + 15) / 16;
  const int gemmBlocks = cdiv_i((long)mtiles * 16 /*256/16 n-tiles*/, 8);
  const int edgeWaveBlocks = cdiv_i(Etot, 8);
  const int ehBlocks = cdiv_i(Etot * 4, 256);
  const int RPB = 128;
  const int statBlocks = cdiv_i(N, RPB);

  // =================== Layer 1 ===================
  gemm_dual_wmma_k<<<gemmBlocks, 256, 0, stream>>>(x, Wl1, bl1, Wr1, br1, A, B, N, 128, 256);

  fill_f32_k<<<1024, 256, 0, stream>>>(C, 0.f, (long)N * 256);
  fill_f32_k<<<256, 256, 0, stream>>>(DEN, 0.f, (long)N * 4);
  fill_u32_k<<<256, 256, 0, stream>>>(MX, ENC_NEG_INF, (long)N * 4);
  fill_f32_k<<<1, 256, 0, stream>>>(SUM, 0.f, 512);  // SUM + SQ contiguous

  edge_logits_k<<<edgeWaveBlocks, 256, 0, stream>>>(ei, E, Etot, A, B, att1, LG);
  seg_max_k<<<ehBlocks, 256, 0, stream>>>(ei, E, Etot, LG, MX);
  seg_exp_k<<<ehBlocks, 256, 0, stream>>>(ei, E, Etot, LG, MX, DEN);
  aggregate_k<<<edgeWaveBlocks, 256, 0, stream>>>(ei, E, Etot, A, LG, DEN, C);

  bias_leaky_stats_k<<<statBlocks, 256, 0, stream>>>(C, bias1, SUM, SQ, N, 256, RPB);
  bn_norm_k<<<2048, 256, 0, stream>>>(C, C, SUM, SQ, g1, beta1, N, 256);

  // =================== Layer 2 ===================
  gemm_dual_wmma_k<<<gemmBlocks, 256, 0, stream>>>(C, Wl2, bl2, Wr2, br2, A, B, N, 256, 256);

  fill_f32_k<<<1024, 256, 0, stream>>>(C, 0.f, (long)N * 256);
  fill_f32_k<<<256, 256, 0, stream>>>(DEN, 0.f, (long)N * 4);
  fill_u32_k<<<256, 256, 0, stream>>>(MX, ENC_NEG_INF, (long)N * 4);
  fill_f32_k<<<1, 256, 0, stream>>>(SUM, 0.f, 512);

  edge_logits_k<<<edgeWaveBlocks, 256, 0, stream>>>(ei, E, Etot, A, B, att2, LG);
  seg_max_k<<<ehBlocks, 256, 0, stream>>>(ei, E, Etot, LG, MX);
  seg_exp_k<<<ehBlocks, 256, 0, stream>>>(ei, E, Etot, LG, MX, DEN);
  aggregate_k<<<edgeWaveBlocks, 256, 0, stream>>>(ei, E, Etot, A, LG, DEN, C);

  head_mean_stats_k<<<statBlocks, 64, 0, stream>>>(C, bias2, H2, SUM, SQ, N, RPB);
  bn_norm_k<<<1024, 256, 0, stream>>>(H2, out, SUM, SQ, g2, beta2, N, 64);
}